// logistic_fun_learnable_M_45861660787346
// MI455X (gfx1250) — compile-verified
//
#include <hip/hip_runtime.h>

typedef __attribute__((ext_vector_type(2))) float v2f;
typedef __attribute__((ext_vector_type(8))) float v8f;

#define NPAIR 153   // pairs (b<=c) over 17 indices
#define KPAD  160   // padded K dimension (40 WMMA k-steps of 4)
#define NKC   40
#define GATE_KF 10.0f
#define GATE_OFF 0.5f

// ---- compile-time pair table: j -> (b, c), b<=c<=16, triangular order ----
struct PairTab { int b[KPAD]; int c[KPAD]; };
static constexpr PairTab mkTab() {
  PairTab t{};
  int j = 0;
  for (int bb = 0; bb < 17; ++bb)
    for (int cc = bb; cc < 17; ++cc) { t.b[j] = bb; t.c[j] = cc; ++j; }
  for (; j < KPAD; ++j) { t.b[j] = 0; t.c[j] = 0; }
  return t;
}
static constexpr PairTab TAB = mkTab();

// ---- static product generator: fills A-frag values for this lane half ----
// Lane<16 needs j = 4*kc+{0,1}; lane>=16 needs j = 4*kc+{2,3} (f32 A-matrix
// 16x4 VGPR layout). All xv[] indices are compile-time constants -> stays in
// VGPRs, no scratch.
template<int OFF, int KC>
struct Gen {
  static __device__ __forceinline__ void run(const float* xv, float* af) {
    constexpr int j0 = 4 * KC + OFF;
    constexpr int j1 = j0 + 1;
    constexpr int b0 = TAB.b[j0], c0 = TAB.c[j0];
    constexpr int b1 = TAB.b[j1], c1 = TAB.c[j1];
    af[2 * KC]     = (j0 < NPAIR) ? xv[b0] * xv[c0] : 0.0f;
    af[2 * KC + 1] = (j1 < NPAIR) ? xv[b1] * xv[c1] : 0.0f;
    Gen<OFF, KC + 1>::run(xv, af);
  }
};
template<int OFF> struct Gen<OFF, NKC> {
  static __device__ __forceinline__ void run(const float*, float*) {}
};

__device__ __forceinline__ float sigmoidf_(float z) {
  return 1.0f / (1.0f + __expf(-z));
}

// ---- kernel 1: build gated coefficient matrix C[16][160] in workspace ----
// C[a][pair(b,c)] = w[i] * gate(deg[i]) for each monomial i>=1.
// i==0 (constant term, a==16) is handled as +w[0] in the epilogue.
__global__ __launch_bounds__(1024)
void build_C_kernel(const float* __restrict__ w, const float* __restrict__ M_learned,
                    const int* __restrict__ comb_idx, const int* __restrict__ degrees,
                    float* __restrict__ C, int p) {
  const int tid = threadIdx.x;
  for (int t = tid; t < 16 * KPAD; t += blockDim.x) C[t] = 0.0f;
  __syncthreads();
  const float m = M_learned[0];
  const float m_int = rintf(m);           // forward value of the STE round
  for (int i = tid; i < p; i += blockDim.x) {
    if (i == 0) continue;                 // constant feature: handled as +w[0]
    const int a = comb_idx[3 * i + 0];
    const int b = comb_idx[3 * i + 1];
    const int c = comb_idx[3 * i + 2];
    const int d = degrees[i];
    const float gate = sigmoidf_(GATE_KF * (m_int - (float)d + GATE_OFF));
    // triangular pair index over 17 values
    const int j = b * 17 - (b * (b - 1)) / 2 + (c - b);
    C[a * KPAD + j] = w[i] * gate;        // (a,j) unique per i -> plain store
  }
}

// ---- kernel 2: fused Q-build + H = Q * C^T via v_wmma_f32_16x16x4_f32 ----
// Two independent accumulator chains (even/odd k-chunk) double XDL ILP.
__global__ __launch_bounds__(256)
void poly_wmma_kernel(const float* __restrict__ X, const float* __restrict__ w,
                      const float* __restrict__ Cg, float* __restrict__ out, int N) {
  __shared__ float C_lds[16 * KPAD];       // 10 KB: B-matrix source, shared by block
  __shared__ float H_lds[8][16][17];       // staged D tiles (pad 17 vs bank conflicts)
  __shared__ float x_lds[8][16][17];       // staged x rows for epilogue dot

  for (int t = threadIdx.x; t < 16 * KPAD; t += 256) C_lds[t] = Cg[t];
  __syncthreads();

  const int wave = threadIdx.x >> 5;
  const int lane = threadIdx.x & 31;
  const int m    = lane & 15;              // sample row within tile
  const int tile = blockIdx.x * 8 + wave;  // one 16-sample tile per wave
  int n = tile * 16 + m;
  const int ncl = (n < N) ? n : (N - 1);   // clamped load index

  // load this lane's sample row (both lane halves hold the same row)
  float xv[17];
  {
    const float4* xp = (const float4*)(X + (size_t)ncl * 16);
    float4 r0 = xp[0], r1 = xp[1], r2 = xp[2], r3 = xp[3];
    xv[0] = r0.x;  xv[1] = r0.y;  xv[2] = r0.z;  xv[3] = r0.w;
    xv[4] = r1.x;  xv[5] = r1.y;  xv[6] = r1.z;  xv[7] = r1.w;
    xv[8] = r2.x;  xv[9] = r2.y;  xv[10] = r2.z; xv[11] = r2.w;
    xv[12] = r3.x; xv[13] = r3.y; xv[14] = r3.z; xv[15] = r3.w;
    xv[16] = 1.0f;                        // bias column
  }

  // per-lane A-fragment values (Q tile), fully static register indexing
  float af[2 * NKC];
  if (lane < 16) Gen<0, 0>::run(xv, af);
  else           Gen<2, 0>::run(xv, af);

  // B-fragment base: lane supplies column a = lane&15, K = 4*kc + (0|2) + {0,1}
  const float* bp = &C_lds[(lane & 15) * KPAD + ((lane < 16) ? 0 : 2)];

  // H[m, 0..15] over K = 160: 40 f32 WMMAs in two interleaved chains
  v8f acc0 = {};
  v8f acc1 = {};
#pragma unroll
  for (int kp = 0; kp < NKC / 2; ++kp) {
    const int kc0 = 2 * kp, kc1 = 2 * kp + 1;
    // one 2ADDR-fusable pair of b64 LDS reads feeds both chains
    float2 bv0 = *reinterpret_cast<const float2*>(bp + 4 * kc0);
    float2 bv1 = *reinterpret_cast<const float2*>(bp + 4 * kc1);
    v2f a0; a0.x = af[2 * kc0]; a0.y = af[2 * kc0 + 1];
    v2f a1; a1.x = af[2 * kc1]; a1.y = af[2 * kc1 + 1];
    v2f b0; b0.x = bv0.x; b0.y = bv0.y;
    v2f b1; b1.x = bv1.x; b1.y = bv1.y;
    acc0 = __builtin_amdgcn_wmma_f32_16x16x4_f32(
        false, a0, false, b0, (short)0, acc0, false, false);
    acc1 = __builtin_amdgcn_wmma_f32_16x16x4_f32(
        false, a1, false, b1, (short)0, acc1, false, false);
  }
  v8f acc = acc0 + acc1;

  // stage D tile: VGPR r -> row (r | r+8), col = lane&15 (ISA C/D layout)
  const int rowbase = (lane < 16) ? 0 : 8;
#pragma unroll
  for (int r = 0; r < 8; ++r) H_lds[wave][rowbase + r][m] = acc[r];
  if (lane < 16) {
#pragma unroll
    for (int a = 0; a < 16; ++a) x_lds[wave][m][a] = xv[a];
  }
  __syncthreads();

  // epilogue: logit[n] = w0 + sum_a x[n,a] * H[n,a] ; out = sigmoid(logit)
  if (lane < 16 && n < N) {
    float z = w[0];
#pragma unroll
    for (int a = 0; a < 16; ++a) z += x_lds[wave][lane][a] * H_lds[wave][lane][a];
    out[n] = sigmoidf_(z);
  }
}

extern "C" void kernel_launch(void* const* d_in, const int* in_sizes, int n_in,
                              void* d_out, int out_size, void* d_ws, size_t ws_size,
                              hipStream_t stream) {
  const float* X  = (const float*)d_in[0];
  const float* w  = (const float*)d_in[1];
  const float* Ml = (const float*)d_in[2];
  const int* comb = (const int*)d_in[3];
  const int* degs = (const int*)d_in[4];
  float* out = (float*)d_out;
  float* Cg  = (float*)d_ws;               // 16*160*4 = 10240 bytes of scratch

  const int p = in_sizes[1];
  const int N = in_sizes[0] / 16;

  build_C_kernel<<<1, 1024, 0, stream>>>(w, Ml, comb, degs, Cg, p);
  const int blocks = (N + 127) / 128;      // 8 waves/block, 16 samples/wave
  poly_wmma_kernel<<<blocks, 256, 0, stream>>>(X, w, Cg, out, N);
}